// ListMLELoss_36532991819995
// MI455X (gfx1250) — compile-verified
//
#include <hip/hip_runtime.h>
#include <stdint.h>

// ---------------------------------------------------------------------------
// ListMLE loss, MI455X (gfx1250, wave32).
//   lse_suffix_i = m + log( sum_{j: t_j <= t_i} exp(s_j - m) )
// => sort ascending by target key, inclusive prefix scan of exp(s-m), then
//    loss = -mean( log( exp(s_i - lse_i) + 1e-10 ) ).
// Memory-bound: 4-pass stable LSD radix sort (8-bit digits) on (key, s) pairs
// with async global->LDS tile staging + LDS-reordered coalesced async
// LDS->global output, then a chained tile scan + fused loss. Final reduction
// uses V_WMMA_F32_16X16X4_F32 (ones-B => exact f32 row-sum accumulate).
// Workspace assumption: ws_size >= ~260 MB for N = 16.7M.
// ---------------------------------------------------------------------------

#define TILE   4096
#define BT     256
#define ROUNDS 16      // TILE / BT
#define RADIX  256
#define WAVES  8       // BT / 32

typedef __attribute__((ext_vector_type(2))) float v2f;
typedef __attribute__((ext_vector_type(8))) float v8f;

__device__ __forceinline__ uint32_t orderKey(float t) {
  uint32_t b = __float_as_uint(t);
  // monotone map: ascending uint  <=>  ascending float
  return (b & 0x80000000u) ? ~b : (b | 0x80000000u);
}

template <typename T>
__device__ __forceinline__ T waveIncScan(T v, int lane) {
  #pragma unroll
  for (int o = 1; o < 32; o <<= 1) {
    T n = __shfl_up(v, o, 32);
    if (lane >= o) v += n;
  }
  return v;
}

// 256-thread block exclusive scan (8 waves). lds8 must hold WAVES elements.
template <typename T>
__device__ __forceinline__ T blockExScan256(T v, T* lds8) {
  const int lane = threadIdx.x & 31;
  const int wv   = threadIdx.x >> 5;
  T inc = waveIncScan<T>(v, lane);
  if (lane == 31) lds8[wv] = inc;
  __syncthreads();
  if (wv == 0) {
    T t = (lane < WAVES) ? lds8[lane] : (T)0;
    T ti = waveIncScan<T>(t, lane);
    if (lane < WAVES) lds8[lane] = ti - t;   // exclusive wave base
  }
  __syncthreads();
  T r = lds8[wv] + inc - v;
  __syncthreads();
  return r;
}

// ---------------------- max reduction ----------------------
__global__ void __launch_bounds__(BT) kmax1(const float* __restrict__ x,
                                            uint32_t n, float* __restrict__ part) {
  float m = -3.402823466e38f;
  const float4* x4 = (const float4*)x;
  uint32_t qbase = (uint32_t)blockIdx.x * (TILE / 4);
  #pragma unroll
  for (int r = 0; r < 4; ++r) {
    uint32_t q = qbase + (uint32_t)r * BT + threadIdx.x;
    uint32_t e = q * 4u;
    if (e + 3u < n) {
      float4 v = x4[q];
      m = fmaxf(m, fmaxf(fmaxf(v.x, v.y), fmaxf(v.z, v.w)));
    } else {
      for (int j = 0; j < 4; ++j)
        if (e + j < n) m = fmaxf(m, x[e + j]);
    }
  }
  for (int o = 16; o > 0; o >>= 1) m = fmaxf(m, __shfl_xor(m, o, 32));
  __shared__ float lds[WAVES];
  if ((threadIdx.x & 31) == 0) lds[threadIdx.x >> 5] = m;
  __syncthreads();
  if (threadIdx.x == 0) {
    float t = lds[0];
    for (int i = 1; i < WAVES; ++i) t = fmaxf(t, lds[i]);
    part[blockIdx.x] = t;
  }
}

__global__ void __launch_bounds__(BT) kmax2(const float* __restrict__ part,
                                            uint32_t n, float* __restrict__ out) {
  float m = -3.402823466e38f;
  for (uint32_t i = threadIdx.x; i < n; i += BT) m = fmaxf(m, part[i]);
  for (int o = 16; o > 0; o >>= 1) m = fmaxf(m, __shfl_xor(m, o, 32));
  __shared__ float lds[WAVES];
  if ((threadIdx.x & 31) == 0) lds[threadIdx.x >> 5] = m;
  __syncthreads();
  if (threadIdx.x == 0) {
    float t = lds[0];
    for (int i = 1; i < WAVES; ++i) t = fmaxf(t, lds[i]);
    out[0] = t;
  }
}

// ---------------------- radix sort: per-tile histogram ----------------------
__global__ void __launch_bounds__(BT) khist(const float* __restrict__ targ,
                                            const uint32_t* __restrict__ srcK,
                                            int raw, uint32_t n, int shift,
                                            uint32_t* __restrict__ hist,
                                            uint32_t nTiles) {
  __shared__ uint32_t h[RADIX];
  h[threadIdx.x] = 0;
  __syncthreads();
  uint32_t qbase = (uint32_t)blockIdx.x * (TILE / 4);
  #pragma unroll
  for (int r = 0; r < 4; ++r) {
    uint32_t q = qbase + (uint32_t)r * BT + threadIdx.x;
    uint32_t e = q * 4u;
    if (e + 3u < n) {
      uint32_t k0, k1, k2, k3;
      if (raw) {
        float4 t = ((const float4*)targ)[q];
        k0 = orderKey(t.x); k1 = orderKey(t.y); k2 = orderKey(t.z); k3 = orderKey(t.w);
      } else {
        uint4 k = ((const uint4*)srcK)[q];
        k0 = k.x; k1 = k.y; k2 = k.z; k3 = k.w;
      }
      atomicAdd(&h[(k0 >> shift) & (RADIX - 1)], 1u);
      atomicAdd(&h[(k1 >> shift) & (RADIX - 1)], 1u);
      atomicAdd(&h[(k2 >> shift) & (RADIX - 1)], 1u);
      atomicAdd(&h[(k3 >> shift) & (RADIX - 1)], 1u);
    } else {
      for (int j = 0; j < 4; ++j)
        if (e + j < n) {
          uint32_t k = raw ? orderKey(targ[e + j]) : srcK[e + j];
          atomicAdd(&h[(k >> shift) & (RADIX - 1)], 1u);
        }
    }
  }
  __syncthreads();
  hist[(size_t)threadIdx.x * nTiles + blockIdx.x] = h[threadIdx.x];  // digit-major
}

// ---------------------- radix sort: hist table scan ----------------------
__global__ void __launch_bounds__(BT) krowsum(const uint32_t* __restrict__ hist,
                                              uint32_t nTiles,
                                              uint32_t* __restrict__ dTot) {
  const uint32_t* row = hist + (size_t)blockIdx.x * nTiles;
  uint32_t s = 0;
  for (uint32_t i = threadIdx.x; i < nTiles; i += BT) s += row[i];
  for (int o = 16; o > 0; o >>= 1) s += __shfl_xor(s, o, 32);
  __shared__ uint32_t lds[WAVES];
  if ((threadIdx.x & 31) == 0) lds[threadIdx.x >> 5] = s;
  __syncthreads();
  if (threadIdx.x == 0) {
    uint32_t t = 0;
    for (int i = 0; i < WAVES; ++i) t += lds[i];
    dTot[blockIdx.x] = t;
  }
}

__global__ void __launch_bounds__(BT) kdigitscan(const uint32_t* __restrict__ dTot,
                                                 uint32_t* __restrict__ dBase) {
  __shared__ uint32_t lds8[WAVES];
  uint32_t v = dTot[threadIdx.x];
  dBase[threadIdx.x] = blockExScan256<uint32_t>(v, lds8);
}

__global__ void __launch_bounds__(BT) krowscan(uint32_t* __restrict__ hist,
                                               uint32_t nTiles,
                                               const uint32_t* __restrict__ dBase) {
  __shared__ uint32_t lds8[WAVES];
  uint32_t* row = hist + (size_t)blockIdx.x * nTiles;
  uint32_t per = (nTiles + BT - 1) / BT;
  uint32_t start = threadIdx.x * per;
  uint32_t tot = 0;
  for (uint32_t j = 0; j < per; ++j) {
    uint32_t i = start + j;
    if (i < nTiles) tot += row[i];
  }
  uint32_t ex = blockExScan256<uint32_t>(tot, lds8);
  uint32_t run = dBase[blockIdx.x] + ex;
  for (uint32_t j = 0; j < per; ++j) {
    uint32_t i = start + j;
    if (i < nTiles) { uint32_t t = row[i]; row[i] = run; run += t; }
  }
}

// ---------------------- radix sort: stable scatter ----------------------
// Phase A: async-stage tile to LDS, rank all rounds (regs), count per digit.
// Phase B: scan counts, reorder (key,val) into LDS tile-sorted order, then
//          emit coalesced async LDS->global stores. ASYNCcnt drains at endpgm.
__global__ void __launch_bounds__(BT) kscatter(const float* __restrict__ targ,
                                               const float* __restrict__ pred,
                                               const uint32_t* __restrict__ srcK,
                                               const float* __restrict__ srcV,
                                               int raw, int writeKeys,
                                               uint32_t n, int shift,
                                               const uint32_t* __restrict__ histBase,
                                               uint32_t nTiles,
                                               uint32_t* __restrict__ dstK,
                                               float* __restrict__ dstV) {
  __shared__ uint32_t gbase[RADIX];
  __shared__ uint32_t running[RADIX];
  __shared__ uint32_t waveHist[WAVES][RADIX];
  __shared__ uint32_t lds8[WAVES];
  __shared__ __align__(16) uint32_t ldsKey[TILE];
  __shared__ __align__(16) float    ldsVal[TILE];

  const int tid = threadIdx.x;
  const int lane = tid & 31;
  const int wv = tid >> 5;
  const uint32_t lt = (1u << lane) - 1u;
  const uint32_t tbase = (uint32_t)blockIdx.x * TILE;
  const uint32_t tileCnt = min((uint32_t)TILE, n - tbase);

  gbase[tid] = histBase[(size_t)tid * nTiles + blockIdx.x];
  running[tid] = 0;

  const uint64_t keySrc = (uint64_t)(uintptr_t)(raw ? (const void*)targ : (const void*)srcK);
  const uint64_t valSrc = (uint64_t)(uintptr_t)(raw ? (const void*)pred : (const void*)srcV);
  const uint32_t ldsKeyBase = (uint32_t)(uintptr_t)&ldsKey[0];
  const uint32_t ldsValBase = (uint32_t)(uintptr_t)&ldsVal[0];

  if (tileCnt == TILE) {
    // Stage 32KB tile via async global->LDS (16B per lane per issue).
    #pragma unroll
    for (int q = 0; q < 4; ++q) {
      uint32_t j0 = ((uint32_t)q * BT + tid) * 4u;      // element index in tile
      uint32_t goff = (tbase + j0) * 4u;                // byte offset
      asm volatile("global_load_async_to_lds_b128 %0, %1, %2"
                   :: "v"(ldsKeyBase + j0 * 4u), "v"(goff), "s"(keySrc) : "memory");
      asm volatile("global_load_async_to_lds_b128 %0, %1, %2"
                   :: "v"(ldsValBase + j0 * 4u), "v"(goff), "s"(valSrc) : "memory");
    }
    asm volatile("s_wait_asynccnt 0x0" ::: "memory");
  } else {
    for (int r = 0; r < ROUNDS; ++r) {
      uint32_t j = (uint32_t)r * BT + tid;
      if (j < tileCnt) {
        ldsKey[j] = raw ? __float_as_uint(targ[tbase + j]) : srcK[tbase + j];
        ldsVal[j] = raw ? pred[tbase + j] : srcV[tbase + j];
      }
    }
  }
  __syncthreads();

  uint32_t myKey[ROUNDS];
  float    myVal[ROUNDS];
  uint32_t myRank[ROUNDS];

  for (int r = 0; r < ROUNDS; ++r) {
    for (int i = tid; i < WAVES * RADIX; i += BT) ((uint32_t*)waveHist)[i] = 0;
    __syncthreads();

    uint32_t j = (uint32_t)r * BT + tid;
    bool valid = j < tileCnt;
    uint32_t kw = valid ? ldsKey[j] : 0u;
    uint32_t key = raw ? orderKey(__uint_as_float(kw)) : kw;
    float val = valid ? ldsVal[j] : 0.f;
    uint32_t digit = (key >> shift) & (RADIX - 1);

    // wave32 peer matching over the 8 digit bits
    uint32_t peers = (uint32_t)__ballot(valid);
    #pragma unroll
    for (int b = 0; b < 8; ++b) {
      bool bit = (digit >> b) & 1;
      uint32_t bal = (uint32_t)__ballot(valid && bit);
      peers &= bit ? bal : ~bal;
    }
    uint32_t lrank = __popc(peers & lt);
    if (valid && (peers & lt) == 0u)          // peer-group leader
      waveHist[wv][digit] = __popc(peers);
    __syncthreads();

    uint32_t pre = 0;
    for (int w2 = 0; w2 < wv; ++w2) pre += waveHist[w2][digit];
    myKey[r]  = key;
    myVal[r]  = val;
    myRank[r] = running[digit] + pre + lrank;  // tile-local rank within digit
    __syncthreads();

    uint32_t tot = 0;
    #pragma unroll
    for (int w2 = 0; w2 < WAVES; ++w2) tot += waveHist[w2][tid];
    running[tid] += tot;
    __syncthreads();
  }

  // running[d] == per-digit tile count; exclusive-scan -> tile start offsets.
  uint32_t cnt = running[tid];
  uint32_t tstart = blockExScan256<uint32_t>(cnt, lds8);
  running[tid] = tstart;                        // reuse as tileStart
  __syncthreads();

  // Reorder into LDS tile-sorted order.
  for (int r = 0; r < ROUNDS; ++r) {
    uint32_t j = (uint32_t)r * BT + tid;
    if (j < tileCnt) {
      uint32_t d = (myKey[r] >> shift) & (RADIX - 1);
      uint32_t p = running[d] + myRank[r];
      ldsKey[p] = myKey[r];
      ldsVal[p] = myVal[r];
    }
  }
  __syncthreads();

  // Coalesced output straight from LDS (no VGPR round-trip).
  const uint64_t dstK64 = (uint64_t)(uintptr_t)dstK;
  const uint64_t dstV64 = (uint64_t)(uintptr_t)dstV;
  for (int r = 0; r < ROUNDS; ++r) {
    uint32_t j = (uint32_t)r * BT + tid;
    if (j < tileCnt) {
      uint32_t k = ldsKey[j];
      uint32_t d = (k >> shift) & (RADIX - 1);
      uint32_t pos = gbase[d] + (j - running[d]);
      if (writeKeys)
        asm volatile("global_store_async_from_lds_b32 %0, %1, %2"
                     :: "v"(pos * 4u), "v"(ldsKeyBase + j * 4u), "s"(dstK64) : "memory");
      asm volatile("global_store_async_from_lds_b32 %0, %1, %2"
                   :: "v"(pos * 4u), "v"(ldsValBase + j * 4u), "s"(dstV64) : "memory");
    }
  }
  // implicit S_WAIT_IDLE at S_ENDPGM drains ASYNCcnt before LDS is released
}

// ---------------------- scan of exp(s-m) + fused loss ----------------------
__global__ void __launch_bounds__(BT) kblocksum(const float* __restrict__ s,
                                                const float* __restrict__ mPtr,
                                                uint32_t n, float* __restrict__ bs) {
  const float m = *mPtr;
  const float4* s4 = (const float4*)s;
  float acc = 0.f;
  uint32_t qbase = (uint32_t)blockIdx.x * (TILE / 4);
  #pragma unroll
  for (int r = 0; r < 4; ++r) {
    uint32_t q = qbase + (uint32_t)r * BT + threadIdx.x;
    uint32_t e = q * 4u;
    if (e + 3u < n) {
      float4 v = s4[q];
      acc += expf(v.x - m) + expf(v.y - m) + expf(v.z - m) + expf(v.w - m);
    } else {
      for (int j = 0; j < 4; ++j)
        if (e + j < n) acc += expf(s[e + j] - m);
    }
  }
  for (int o = 16; o > 0; o >>= 1) acc += __shfl_xor(acc, o, 32);
  __shared__ float lds[WAVES];
  if ((threadIdx.x & 31) == 0) lds[threadIdx.x >> 5] = acc;
  __syncthreads();
  if (threadIdx.x == 0) {
    float t = 0.f;
    for (int i = 0; i < WAVES; ++i) t += lds[i];
    bs[blockIdx.x] = t;
  }
}

__global__ void __launch_bounds__(BT) kbase(const float* __restrict__ bs,
                                            uint32_t nTiles, float* __restrict__ bb) {
  __shared__ float lds8[WAVES];
  uint32_t per = (nTiles + BT - 1) / BT;
  uint32_t start = threadIdx.x * per;
  float tot = 0.f;
  for (uint32_t j = 0; j < per; ++j) {
    uint32_t i = start + j;
    if (i < nTiles) tot += bs[i];
  }
  float ex = blockExScan256<float>(tot, lds8);
  float run = ex;
  for (uint32_t j = 0; j < per; ++j) {
    uint32_t i = start + j;
    if (i < nTiles) { bb[i] = run; run += bs[i]; }
  }
}

__global__ void __launch_bounds__(BT) kloss(const float* __restrict__ s,
                                            const float* __restrict__ mPtr,
                                            const float* __restrict__ bb,
                                            uint32_t n, float* __restrict__ pl) {
  __shared__ float lds8[WAVES];
  const float m = *mPtr;
  const float4* s4 = (const float4*)s;
  uint32_t start = (uint32_t)blockIdx.x * TILE + (uint32_t)threadIdx.x * 16u;
  float f[16];
  float tot = 0.f;
  #pragma unroll
  for (int jj = 0; jj < 4; ++jj) {
    uint32_t e = start + (uint32_t)jj * 4u;
    if (e + 3u < n) {
      float4 v = s4[start / 4u + jj];
      f[jj*4+0] = expf(v.x - m); f[jj*4+1] = expf(v.y - m);
      f[jj*4+2] = expf(v.z - m); f[jj*4+3] = expf(v.w - m);
    } else {
      for (int j = 0; j < 4; ++j)
        f[jj*4+j] = (e + j < n) ? expf(s[e + j] - m) : 0.f;
    }
    tot += f[jj*4+0] + f[jj*4+1] + f[jj*4+2] + f[jj*4+3];
  }
  float ex = blockExScan256<float>(tot, lds8);
  float cum = bb[blockIdx.x] + ex;
  float lsum = 0.f;
  #pragma unroll
  for (int j = 0; j < 16; ++j) {
    uint32_t i = start + j;
    cum += f[j];
    if (i < n) lsum += logf(f[j] / cum + 1e-10f);   // log(exp(s-lse)+eps)
  }
  for (int o = 16; o > 0; o >>= 1) lsum += __shfl_xor(lsum, o, 32);
  __shared__ float ldsr[WAVES];
  if ((threadIdx.x & 31) == 0) ldsr[threadIdx.x >> 5] = lsum;
  __syncthreads();
  if (threadIdx.x == 0) {
    float t = 0.f;
    for (int i = 0; i < WAVES; ++i) t += ldsr[i];
    pl[blockIdx.x] = t;
  }
}

// ---------------------- WMMA final reduction (one wave) ----------------------
// D = A(16x4) x B(4x16 ones) + C : every A element contributes exactly once,
// C accumulates in f32 across iterations -> order-free f32 reduction.
__global__ void __launch_bounds__(32) kfinal(const float* __restrict__ part,
                                             uint32_t nTiles, uint32_t n,
                                             float* __restrict__ out) {
  const int lane = threadIdx.x;         // 32 threads, EXEC all ones at wmma
  v8f acc = {0.f, 0.f, 0.f, 0.f, 0.f, 0.f, 0.f, 0.f};
  v2f ones; ones.x = 1.f; ones.y = 1.f;
  const v2f* p2 = (const v2f*)part;
  uint32_t full = nTiles / 64u;
  for (uint32_t it = 0; it < full; ++it) {
    v2f a = p2[it * 32u + (uint32_t)lane];
    acc = __builtin_amdgcn_wmma_f32_16x16x4_f32(false, a, false, ones,
                                                (short)0, acc, false, false);
  }
  if (full * 64u < nTiles) {            // uniform-branch tail
    uint32_t i0 = full * 64u + (uint32_t)lane * 2u;
    v2f a;
    a.x = (i0 < nTiles) ? part[i0] : 0.f;
    a.y = (i0 + 1u < nTiles) ? part[i0 + 1u] : 0.f;
    acc = __builtin_amdgcn_wmma_f32_16x16x4_f32(false, a, false, ones,
                                                (short)0, acc, false, false);
  }
  // D rows 0..7 live in lane 0's 8 VGPRs, rows 8..15 in lane 16's.
  float srow = acc[0] + acc[1] + acc[2] + acc[3] + acc[4] + acc[5] + acc[6] + acc[7];
  float total = __shfl(srow, 0, 32) + __shfl(srow, 16, 32);
  if (lane == 0) out[0] = -total / (float)n;
}

// ---------------------------------------------------------------------------
extern "C" void kernel_launch(void* const* d_in, const int* in_sizes, int n_in,
                              void* d_out, int out_size, void* d_ws, size_t ws_size,
                              hipStream_t stream) {
  const float* preds = (const float*)d_in[0];
  const float* targs = (const float*)d_in[1];
  uint32_t N = (uint32_t)in_sizes[0];
  uint32_t nTiles = (N + TILE - 1) / TILE;

  char* w = (char*)d_ws;
  auto alloc = [&](size_t bytes) -> char* {
    char* p = w;
    w += (bytes + 255) & ~(size_t)255;
    return p;
  };
  uint32_t* keysA = (uint32_t*)alloc((size_t)N * 4);
  float*    valsA = (float*)   alloc((size_t)N * 4);
  uint32_t* keysB = (uint32_t*)alloc((size_t)N * 4);
  float*    valsB = (float*)   alloc((size_t)N * 4);
  uint32_t* hist  = (uint32_t*)alloc((size_t)RADIX * nTiles * 4);
  uint32_t* dTot  = (uint32_t*)alloc(RADIX * 4);
  uint32_t* dBase = (uint32_t*)alloc(RADIX * 4);
  float* blockSums = (float*)alloc((size_t)nTiles * 4);
  float* blockBase = (float*)alloc((size_t)nTiles * 4);
  float* partLoss  = (float*)alloc((size_t)nTiles * 4);
  float* maxPart   = (float*)alloc((size_t)nTiles * 4);
  float* mPtr      = (float*)alloc(4);

  dim3 g(nTiles), b(BT);

  kmax1<<<g, b, 0, stream>>>(preds, N, maxPart);
  kmax2<<<1, b, 0, stream>>>(maxPart, nTiles, mPtr);

  const uint32_t* sk[4] = { nullptr, keysA, keysB, keysA };
  const float*    sv[4] = { nullptr, valsA, valsB, valsA };
  uint32_t*       dk[4] = { keysA, keysB, keysA, keysB };
  float*          dv[4] = { valsA, valsB, valsA, valsB };
  for (int p = 0; p < 4; ++p) {
    int shift = 8 * p;
    int raw = (p == 0) ? 1 : 0;
    int writeKeys = (p < 3) ? 1 : 0;   // last pass: values only
    khist<<<g, b, 0, stream>>>(targs, sk[p], raw, N, shift, hist, nTiles);
    krowsum<<<RADIX, b, 0, stream>>>(hist, nTiles, dTot);
    kdigitscan<<<1, b, 0, stream>>>(dTot, dBase);
    krowscan<<<RADIX, b, 0, stream>>>(hist, nTiles, dBase);
    kscatter<<<g, b, 0, stream>>>(targs, preds, sk[p], sv[p], raw, writeKeys,
                                  N, shift, hist, nTiles, dk[p], dv[p]);
  }
  const float* sortedS = valsB;   // ascending by target key after 4 passes

  kblocksum<<<g, b, 0, stream>>>(sortedS, mPtr, N, blockSums);
  kbase<<<1, b, 0, stream>>>(blockSums, nTiles, blockBase);
  kloss<<<g, b, 0, stream>>>(sortedS, mPtr, blockBase, N, partLoss);
  kfinal<<<1, 32, 0, stream>>>(partLoss, nTiles, N, (float*)d_out);
}